// MultivariateGaussianMixture_2662879723707
// MI455X (gfx1250) — compile-verified
//
#include <hip/hip_runtime.h>
#include <hip/hip_bf16.h>
#include <math.h>

typedef __attribute__((ext_vector_type(2))) float v2f;
typedef __attribute__((ext_vector_type(8))) float v8f;

#define NCLUST 16
#define NDIM 8
#define WSTRIDE 80   // padded row stride (floats) for per-cluster coefficient rows
#define NCHUNK 19    // K = 76 = 19 * 4 (WMMA f32 16x16x4)

// ---------------------------------------------------------------------------
// Feature map Phi(x), length 76 (compile-time indexed so x[] stays in VGPRs):
//   0..7   : x[i]                (linear term, coef = Sigma^-1 mu)
//   8      : 1.0                 (constant term, coef = folded const + log w)
//   9..11  : 0.0                 (pad to multiple of 4)
//   12..75 : x[(p)/8]*x[(p)%8]   (full outer product, coef = -0.5 * Sigma^-1)
// Then  ll[n,k] + logw[k] = Phi(x_n) . w_k   -> a [16 x 76] @ [76 x 16] GEMM.
// ---------------------------------------------------------------------------
template <int IDX>
__device__ __forceinline__ float phi_c(const float (&x)[8]) {
  if constexpr (IDX < 8) {
    return x[IDX];
  } else if constexpr (IDX == 8) {
    return 1.0f;
  } else if constexpr (IDX < 12) {
    return 0.0f;
  } else {
    constexpr int p = IDX - 12;
    return x[p >> 3] * x[p & 7];
  }
}

// Build A-operand chunks for one lane-half H.
// ISA layout (32-bit A 16x4): lanes 0-15: VGPR0=K0,VGPR1=K1; lanes 16-31: K2,K3.
template <int H, int T>
struct BuildA {
  static __device__ __forceinline__ void go(v2f (&aa)[NCHUNK], const float (&x)[8]) {
    aa[T].x = phi_c<4 * T + 2 * H>(x);
    aa[T].y = phi_c<4 * T + 2 * H + 1>(x);
    BuildA<H, T + 1>::go(aa, x);
  }
};
template <int H>
struct BuildA<H, NCHUNK> {
  static __device__ __forceinline__ void go(v2f (&)[NCHUNK], const float (&)[8]) {}
};

// ---------------------------------------------------------------------------
// Setup: per cluster k, compute Sigma = tril(chol)tril(chol)^T + I, its
// Cholesky C, logdet, Sigma^-1 = C^-T C^-1, log-softmax(weights), and pack
// the 76-long coefficient vector w_k into W[k*WSTRIDE ...].
// ---------------------------------------------------------------------------
__global__ void gmm_setup(const float* __restrict__ chol, const float* __restrict__ mu,
                          const float* __restrict__ wts, float* __restrict__ W) {
  const int k = threadIdx.x;
  if (k >= NCLUST) return;

  float L[8][8];
  for (int i = 0; i < 8; ++i)
    for (int j = 0; j < 8; ++j)
      L[i][j] = (j <= i) ? chol[(k * 8 + i) * 8 + j] : 0.0f;

  float S[8][8];
  for (int i = 0; i < 8; ++i)
    for (int j = 0; j < 8; ++j) {
      float s = 0.0f;
      for (int p = 0; p < 8; ++p) s += L[i][p] * L[j][p];
      S[i][j] = s + ((i == j) ? 1.0f : 0.0f);
    }

  // Cholesky: S = C C^T (C lower)
  float C[8][8];
  for (int i = 0; i < 8; ++i)
    for (int j = 0; j < 8; ++j) C[i][j] = 0.0f;
  for (int j = 0; j < 8; ++j) {
    float s = S[j][j];
    for (int p = 0; p < j; ++p) s -= C[j][p] * C[j][p];
    const float cjj = sqrtf(s);
    C[j][j] = cjj;
    const float inv = 1.0f / cjj;
    for (int i = j + 1; i < 8; ++i) {
      float t = S[i][j];
      for (int p = 0; p < j; ++p) t -= C[i][p] * C[j][p];
      C[i][j] = t * inv;
    }
  }
  float logdet = 0.0f;
  for (int j = 0; j < 8; ++j) logdet += logf(C[j][j]);
  logdet *= 2.0f;

  // Mi = C^-1 (lower), forward substitution per column
  float Mi[8][8];
  for (int i = 0; i < 8; ++i)
    for (int j = 0; j < 8; ++j) Mi[i][j] = 0.0f;
  for (int c = 0; c < 8; ++c) {
    for (int i = c; i < 8; ++i) {
      float t = (i == c) ? 1.0f : 0.0f;
      for (int p = c; p < i; ++p) t -= C[i][p] * Mi[p][c];
      Mi[i][c] = t / C[i][i];
    }
  }

  // A = Sigma^-1 = Mi^T Mi
  float A[8][8];
  for (int i = 0; i < 8; ++i)
    for (int j = 0; j < 8; ++j) {
      float s = 0.0f;
      for (int p = 0; p < 8; ++p) s += Mi[p][i] * Mi[p][j];
      A[i][j] = s;
    }

  float muk[8];
  for (int i = 0; i < 8; ++i) muk[i] = mu[k * 8 + i];
  float cv[8];  // A @ mu
  float d = 0.0f;
  for (int i = 0; i < 8; ++i) {
    float s = 0.0f;
    for (int j = 0; j < 8; ++j) s += A[i][j] * muk[j];
    cv[i] = s;
    d += s * muk[i];
  }

  // log-softmax of weights (redundant across threads; trivial)
  float mx = wts[0];
  for (int i = 1; i < NCLUST; ++i) mx = fmaxf(mx, wts[i]);
  float se = 0.0f;
  for (int i = 0; i < NCLUST; ++i) se += expf(wts[i] - mx);
  const float lw = wts[k] - mx - logf(se);

  const float LOG2PI = 1.837877066409345339082f;
  float* w = W + k * WSTRIDE;
  for (int i = 0; i < 8; ++i) w[i] = cv[i];                      // +c . x
  w[8] = -0.5f * (d + 8.0f * LOG2PI + logdet) + lw;              // constant
  w[9] = 0.0f; w[10] = 0.0f; w[11] = 0.0f;
  for (int i = 0; i < 8; ++i)
    for (int j = 0; j < 8; ++j)
      w[12 + 8 * i + j] = -0.5f * A[i][j];                       // -0.5 x^T A x
  w[76] = 0.0f; w[77] = 0.0f; w[78] = 0.0f; w[79] = 0.0f;
}

// ---------------------------------------------------------------------------
// Main: each wave processes 16-sample tiles.  D[m][k] = ll + logw comes out of
// 19 V_WMMA_F32_16X16X4_F32 ops split over two independent accumulator chains
// (halved dependence depth); logsumexp over k is a 16-lane butterfly per
// accumulator VGPR.
// ---------------------------------------------------------------------------
__global__ void gmm_main(const float* __restrict__ X, const float* __restrict__ W,
                         float* __restrict__ partials, int ntiles) {
  const int lane  = threadIdx.x & 31;
  const int h     = lane >> 4;   // lane half (selects K-pair inside each chunk)
  const int idx16 = lane & 15;   // A: sample row; B/D: cluster column

  // B operands: per-lane coefficients of cluster idx16 (loaded once, reused).
  // ISA layout (32-bit B 4x16): lanes 0-15: VGPR0=K0,VGPR1=K1; lanes 16-31: K2,K3.
  const float* wrow = W + idx16 * WSTRIDE;
  v2f bb[NCHUNK];
#pragma unroll
  for (int t = 0; t < NCHUNK; ++t)
    bb[t] = *reinterpret_cast<const v2f*>(wrow + 4 * t + 2 * h);  // 8B-aligned

  const int gwave  = (int)((blockIdx.x * blockDim.x + threadIdx.x) >> 5);
  const int nwaves = (int)((gridDim.x * blockDim.x) >> 5);

  float local = 0.0f;

  for (int tile = gwave; tile < ntiles; tile += nwaves) {
    const float4* xp =
        reinterpret_cast<const float4*>(X + (size_t)(tile * 16 + idx16) * NDIM);
    const float4 x0 = xp[0];
    const float4 x1 = xp[1];

    // Prefetch next tile's rows for this lane (clamped -> always valid addr).
    {
      int nt = tile + nwaves;
      nt = (nt < ntiles) ? nt : tile;
      __builtin_prefetch(X + (size_t)(nt * 16 + idx16) * NDIM, 0, 1);
    }

    const float x[8] = {x0.x, x0.y, x0.z, x0.w, x1.x, x1.y, x1.z, x1.w};

    // Divergent only in plain VALU; converged again before the WMMAs.
    v2f aa[NCHUNK];
    if (h == 0) BuildA<0, 0>::go(aa, x);
    else        BuildA<1, 0>::go(aa, x);

    // Two independent accumulation chains over even/odd chunks.
    v8f acc0 = {};
    v8f acc1 = {};
#pragma unroll
    for (int t = 0; t < NCHUNK; t += 2) {
      acc0 = __builtin_amdgcn_wmma_f32_16x16x4_f32(
          false, aa[t], false, bb[t], (short)0, acc0, false, false);
      if (t + 1 < NCHUNK)
        acc1 = __builtin_amdgcn_wmma_f32_16x16x4_f32(
            false, aa[t + 1], false, bb[t + 1], (short)0, acc1, false, false);
    }
    const v8f acc = acc0 + acc1;

    // acc[v] = (ll + logw)[sample m = v + 8*h][cluster = idx16]
#pragma unroll
    for (int v = 0; v < 8; ++v) {
      const float tv = acc[v];
      float mx = tv;
      mx = fmaxf(mx, __shfl_xor(mx, 1, 32));
      mx = fmaxf(mx, __shfl_xor(mx, 2, 32));
      mx = fmaxf(mx, __shfl_xor(mx, 4, 32));
      mx = fmaxf(mx, __shfl_xor(mx, 8, 32));
      float e = __expf(tv - mx);
      e += __shfl_xor(e, 1, 32);
      e += __shfl_xor(e, 2, 32);
      e += __shfl_xor(e, 4, 32);
      e += __shfl_xor(e, 8, 32);
      const float lse = __logf(e) + mx;   // logsumexp for sample m
      local += (idx16 == 0) ? lse : 0.0f; // count each sample exactly once
    }
  }

  // wave butterfly, then per-block LDS reduce -> deterministic partial
  local += __shfl_xor(local, 16, 32);
  local += __shfl_xor(local, 8, 32);
  local += __shfl_xor(local, 4, 32);
  local += __shfl_xor(local, 2, 32);
  local += __shfl_xor(local, 1, 32);

  __shared__ float sred[8];
  const int wv = threadIdx.x >> 5;
  if ((threadIdx.x & 31) == 0) sred[wv] = local;
  __syncthreads();
  if (threadIdx.x == 0) {
    float bsum = 0.0f;
    for (int i = 0; i < 8; ++i) bsum += sred[i];
    partials[blockIdx.x] = bsum;
  }
}

__global__ void gmm_finalize(const float* __restrict__ partials, int nblocks,
                             float* __restrict__ out, float inv_n_neg) {
  __shared__ float sm[256];
  float s = 0.0f;
  for (int i = threadIdx.x; i < nblocks; i += 256) s += partials[i];
  sm[threadIdx.x] = s;
  __syncthreads();
  for (int off = 128; off > 0; off >>= 1) {
    if ((int)threadIdx.x < off) sm[threadIdx.x] += sm[threadIdx.x + off];
    __syncthreads();
  }
  if (threadIdx.x == 0) out[0] = sm[0] * inv_n_neg;  // = -mean
}

extern "C" void kernel_launch(void* const* d_in, const int* in_sizes, int n_in,
                              void* d_out, int out_size, void* d_ws, size_t ws_size,
                              hipStream_t stream) {
  (void)n_in; (void)out_size; (void)ws_size;
  const float* X    = (const float*)d_in[0];
  const float* mu   = (const float*)d_in[1];
  const float* chol = (const float*)d_in[2];
  const float* wts  = (const float*)d_in[3];
  // d_in[4] = it (unused by the math)

  const int n      = in_sizes[0] / NDIM;  // 1,000,000
  const int ntiles = n / 16;              // divides exactly for 1M

  float* W        = (float*)d_ws;                 // 16 * 80 floats
  float* partials = (float*)d_ws + 4096;          // 1024 floats, 16KB offset

  gmm_setup<<<1, 32, 0, stream>>>(chol, mu, wts, W);

  const int blocks = 1024;
  gmm_main<<<blocks, 256, 0, stream>>>(X, W, partials, ntiles);

  gmm_finalize<<<1, 256, 0, stream>>>(partials, blocks, (float*)d_out,
                                      -1.0f / (float)n);
}